// CrossModalComplementaryUncertaintyModule_88699664597410
// MI455X (gfx1250) — compile-verified
//
#include <hip/hip_runtime.h>

// Problem constants from the reference
#define NQ   40000
#define HH   200
#define WW   200
#define CC   256
#define NHH  8
#define HDIM 32
#define FFND 1024

typedef __attribute__((ext_vector_type(2))) float v2f;
typedef __attribute__((ext_vector_type(8))) float v8f;
typedef __attribute__((ext_vector_type(4))) int   v4i;

// ---------------------------------------------------------------------------
// Async global->LDS staging (CDNA5 GLOBAL_LOAD_ASYNC_TO_LDS_B128/B32, ASYNCcnt).
// b128 signature verified in round 2/3: (v4i AS1*, v4i AS3*, imm, imm).
// ---------------------------------------------------------------------------
#if __has_builtin(__builtin_amdgcn_global_load_async_to_lds_b128)
#define USE_ASYNC 1
#else
#define USE_ASYNC 0
#endif

#if USE_ASYNC && __has_builtin(__builtin_amdgcn_global_load_async_to_lds_b32)
#define USE_ASYNC32 1
#else
#define USE_ASYNC32 0
#endif

typedef __attribute__((address_space(1))) v4i* as1_v4i_p;
typedef __attribute__((address_space(3))) v4i* as3_v4i_p;
typedef __attribute__((address_space(1))) int* as1_i32_p;
typedef __attribute__((address_space(3))) int* as3_i32_p;

__device__ __forceinline__ void copy16(const float* __restrict__ g, float* l)
{
#if USE_ASYNC
    v4i* gg = (v4i*)g;   // drop const + bitcast (generic)
    v4i* ll = (v4i*)l;
    __builtin_amdgcn_global_load_async_to_lds_b128(
        (as1_v4i_p)gg, (as3_v4i_p)ll, 0, 0);
#else
    const float4 t = *(const float4*)g;
    l[0] = t.x; l[1] = t.y; l[2] = t.z; l[3] = t.w;
#endif
}

__device__ __forceinline__ void copy4(const float* __restrict__ g, float* l)
{
#if USE_ASYNC32
    int* gg = (int*)g;
    int* ll = (int*)l;
    __builtin_amdgcn_global_load_async_to_lds_b32(
        (as1_i32_p)gg, (as3_i32_p)ll, 0, 0);
#else
    *l = *g;
#endif
}

template <int N>
__device__ __forceinline__ void wait_async()
{
#if USE_ASYNC
#if __has_builtin(__builtin_amdgcn_s_wait_asynccnt)
    __builtin_amdgcn_s_wait_asynccnt(N);
#else
    asm volatile("s_wait_asynccnt %0" :: "i"(N) : "memory");
#endif
#endif
}

// ---------------------------------------------------------------------------
// GEMM: Y[M,N] = act( X[M,K] @ W[K,N] + bias ) (+ residual R)
// Block tile 64xBN (BN=64 or 32), BK=32, 256 threads = 8 waves (4 in M, 2 in N),
// each wave computes a 16x(16*NT) strip with V_WMMA_F32_16X16X4_F32 (exact fp32).
// LDS double-buffered; W stored pair-interleaved (WsP[k>>1][n][k&1]) so every
// A and B fragment is one 8B-aligned ds_load_b64 into an even VGPR pair.
// ---------------------------------------------------------------------------
template <int BN, bool RELU, bool RESID>
__global__ __launch_bounds__(256) void gemm_wmma(
    const float* __restrict__ X, const float* __restrict__ Wt,
    const float* __restrict__ bias, const float* __restrict__ R,
    float* __restrict__ Y, int M, int N, int K)
{
    constexpr int BM   = 64;
    constexpr int BK   = 32;
    constexpr int XPAD = 36;          // stride 36 dw: A-frag b64 reads conflict-free
    constexpr int WC   = BN + 16;     // (BN+16)*2 dw row stride == 32 mod 64:
                                      // kp+1 row hits opposite 32-bank half
    constexpr int NT   = BN / 32;     // 16-col tiles per wave (1 or 2)
    constexpr int WCOPY = (BK * BN) / 256;          // b32 W copies per thread
    constexpr int NCOPY = (USE_ASYNC ? 2 : 0) + (USE_ASYNC32 ? WCOPY : 0);

    __shared__ float Xs [2][BM][XPAD];
    __shared__ float WsP[2][BK / 2][WC][2];

    const int tid   = threadIdx.x;
    const int lane  = tid & 31;
    const int wave  = tid >> 5;
    const int wm    = wave >> 1;      // 0..3 (M sub-tile)
    const int wn    = wave & 1;       // 0..1 (N sub-tile)
    const int m0    = blockIdx.x * BM;
    const int n0    = blockIdx.y * BN;
    const int lhalf = lane >> 4;      // 0 or 1
    const int l15   = lane & 15;

    v8f acc[NT];
    #pragma unroll
    for (int t = 0; t < NT; ++t) {
        const int col = n0 + wn * 16 * NT + t * 16 + l15;
        const float b0 = bias ? bias[col] : 0.0f;
        #pragma unroll
        for (int r = 0; r < 8; ++r) acc[t][r] = b0;
    }

    auto stage = [&](int k0, int buf) {
        // X tile: 64x32 floats = 512 float4, 2 per thread (coalesced b128)
        #pragma unroll
        for (int i = 0; i < 2; ++i) {
            const int idx = tid + i * 256;      // 0..511
            const int r   = idx >> 3;           // 8 float4 per row
            const int c   = (idx & 7) << 2;
            copy16(X + (size_t)(m0 + r) * K + k0 + c, &Xs[buf][r][c]);
        }
        // W tile: BKxBN floats, pair-interleaved in LDS; consecutive lanes read
        // consecutive n -> coalesced 128B global segments, b32 async copies.
        #pragma unroll
        for (int i = 0; i < WCOPY; ++i) {
            const int idx = tid + i * 256;
            const int k   = idx / BN;
            const int n   = idx % BN;
            copy4(Wt + (size_t)(k0 + k) * N + n0 + n, &WsP[buf][k >> 1][n][k & 1]);
        }
    };

    stage(0, 0);
    const int nc = K >> 5;
    for (int c = 0; c < nc; ++c) {
        const int buf = c & 1;
        if (c + 1 < nc) {
            stage((c + 1) << 5, buf ^ 1);
            wait_async<NCOPY>();     // oldest NCOPY (this chunk's) have landed
        } else {
            wait_async<0>();
        }
        __syncthreads();

        // Preload all fragments (single b64 each), then back-to-back WMMAs.
        v2f af[8];
        v2f bf[NT][8];
        #pragma unroll
        for (int i = 0; i < 8; ++i) {
            const int ka = i * 4 + lhalf * 2;   // lanes 0-15: K+0/1, 16-31: K+2/3
            af[i] = *(const v2f*)&Xs[buf][wm * 16 + l15][ka];
            #pragma unroll
            for (int t = 0; t < NT; ++t) {
                const int cb = wn * 16 * NT + t * 16 + l15;
                bf[t][i] = *(const v2f*)&WsP[buf][i * 2 + lhalf][cb][0];
            }
        }
        #pragma unroll
        for (int i = 0; i < 8; ++i) {
            #pragma unroll
            for (int t = 0; t < NT; ++t) {
                acc[t] = __builtin_amdgcn_wmma_f32_16x16x4_f32(
                    false, af[i], false, bf[t][i], (short)0, acc[t], false, false);
            }
        }
        __syncthreads();   // protect buffer reuse before next stage() overwrites
    }

    // C/D layout: VGPR r, lanes 0-15 -> M=r, lanes 16-31 -> M=r+8, N=l15
    #pragma unroll
    for (int t = 0; t < NT; ++t) {
        const int col = n0 + wn * 16 * NT + t * 16 + l15;
        #pragma unroll
        for (int r = 0; r < 8; ++r) {
            const int row = m0 + wm * 16 + r + lhalf * 8;
            float v = acc[t][r];
            if (RELU)  v = fmaxf(v, 0.0f);
            if (RESID) v += R[(size_t)row * N + col];
            Y[(size_t)row * N + col] = v;
        }
    }
}

// ---------------------------------------------------------------------------
// Deformable sampling: one wave per (query, head); lane = channel in head.
// ---------------------------------------------------------------------------
__device__ __forceinline__ float samp_conf(const float* __restrict__ C, int iy, int ix)
{
    if (iy < 0 || iy >= HH || ix < 0 || ix >= WW) return 0.0f;
    return C[iy * WW + ix];
}

__device__ __forceinline__ float samp_val(const float* __restrict__ V,
                                          int iy, int ix, int h, int lane)
{
    if (iy < 0 || iy >= HH || ix < 0 || ix >= WW) return 0.0f;
    return V[(((size_t)iy * WW + ix) * NHH + h) * HDIM + lane];
}

__global__ __launch_bounds__(256) void deform_sample(
    const float* __restrict__ V,     // [H,W,NH,HD] (== vp output [NQ,C])
    const float* __restrict__ OFF,   // [NQ, 64]  (NH,L=1,P=4,2)
    const float* __restrict__ AWL,   // [NQ, 32]  (NH,L*P=4) logits
    const float* __restrict__ CONF,  // [H,W]
    float* __restrict__ S)           // [NQ, NH, HD] == [NQ, C]
{
    const int lane = threadIdx.x & 31;
    const int wave = blockIdx.x * 8 + (threadIdx.x >> 5);
    const int q = wave >> 3;
    const int h = wave & 7;
    if (q >= NQ) return;
    const int qx = q % WW;
    const int qy = q / WW;

    float lg[4], ox[4], oy[4];
    #pragma unroll
    for (int p = 0; p < 4; ++p) {
        lg[p] = AWL[(size_t)q * 32 + h * 4 + p];
        ox[p] = OFF[(size_t)q * 64 + h * 8 + p * 2 + 0];
        oy[p] = OFF[(size_t)q * 64 + h * 8 + p * 2 + 1];
    }
    // softmax over P
    float mx = fmaxf(fmaxf(lg[0], lg[1]), fmaxf(lg[2], lg[3]));
    float e[4], den = 0.0f;
    #pragma unroll
    for (int p = 0; p < 4; ++p) { e[p] = expf(lg[p] - mx); den += e[p]; }
    const float invden = 1.0f / den;

    int   ix0[4], iy0[4];
    float w00[4], w01[4], w10[4], w11[4], a[4];
    float asum = 0.0f;
    #pragma unroll
    for (int p = 0; p < 4; ++p) {
        const float x  = (float)qx + ox[p];   // ref*W - 0.5 + off simplifies
        const float y  = (float)qy + oy[p];
        const float fx = floorf(x), fy = floorf(y);
        ix0[p] = (int)fx; iy0[p] = (int)fy;
        const float wx = x - fx, wy = y - fy;
        w00[p] = (1.0f - wy) * (1.0f - wx);
        w01[p] = (1.0f - wy) * wx;
        w10[p] = wy * (1.0f - wx);
        w11[p] = wy * wx;
        const float cm =
            samp_conf(CONF, iy0[p],     ix0[p]    ) * w00[p] +
            samp_conf(CONF, iy0[p],     ix0[p] + 1) * w01[p] +
            samp_conf(CONF, iy0[p] + 1, ix0[p]    ) * w10[p] +
            samp_conf(CONF, iy0[p] + 1, ix0[p] + 1) * w11[p];
        a[p] = (e[p] * invden) * cm;
        asum += a[p];
    }
    const float rnorm = 1.0f / fmaxf(asum, 1e-6f);

    float out = 0.0f;
    #pragma unroll
    for (int p = 0; p < 4; ++p) {
        const float s =
            samp_val(V, iy0[p],     ix0[p],     h, lane) * w00[p] +
            samp_val(V, iy0[p],     ix0[p] + 1, h, lane) * w01[p] +
            samp_val(V, iy0[p] + 1, ix0[p],     h, lane) * w10[p] +
            samp_val(V, iy0[p] + 1, ix0[p] + 1, h, lane) * w11[p];
        out += s * (a[p] * rnorm);
    }
    S[((size_t)q * NHH + h) * HDIM + lane] = out;
}

// ---------------------------------------------------------------------------
// LayerNorm over C=256: one wave per row, lanes cover channels lane + 32*j.
// Optional fused residual: y = LN(Xa + Xb).
// ---------------------------------------------------------------------------
__global__ __launch_bounds__(256) void layernorm(
    const float* __restrict__ Xa, const float* __restrict__ Xb,
    const float* __restrict__ g,  const float* __restrict__ b,
    float* __restrict__ Y, int M)
{
    const int lane = threadIdx.x & 31;
    const int row  = blockIdx.x * 8 + (threadIdx.x >> 5);
    if (row >= M) return;

    const size_t base = (size_t)row * CC;
    float v[8];
    float s = 0.0f;
    #pragma unroll
    for (int j = 0; j < 8; ++j) {
        const int c = lane + 32 * j;
        float t = Xa[base + c];
        if (Xb) t += Xb[base + c];
        v[j] = t;
        s += t;
    }
    #pragma unroll
    for (int o = 16; o > 0; o >>= 1) s += __shfl_xor(s, o, 32);
    const float mu = s * (1.0f / CC);

    float var = 0.0f;
    #pragma unroll
    for (int j = 0; j < 8; ++j) { const float d = v[j] - mu; var += d * d; }
    #pragma unroll
    for (int o = 16; o > 0; o >>= 1) var += __shfl_xor(var, o, 32);
    const float rstd = rsqrtf(var * (1.0f / CC) + 1e-5f);

    #pragma unroll
    for (int j = 0; j < 8; ++j) {
        const int c = lane + 32 * j;
        Y[base + c] = (v[j] - mu) * rstd * g[c] + b[c];
    }
}

// ---------------------------------------------------------------------------
// Host-side orchestration
// ---------------------------------------------------------------------------
extern "C" void kernel_launch(void* const* d_in, const int* in_sizes, int n_in,
                              void* d_out, int out_size, void* d_ws, size_t ws_size,
                              hipStream_t stream)
{
    auto F = [&](int i) { return (const float*)d_in[i]; };

    const float* img_feat = F(0);
    const float* pts_feat = F(1);
    // d_in[2] (u_img) and d_in[3] (u_pts) are unused by the reference
    const float* c_img = F(4);
    const float* c_pts = F(5);

    // params flattened in JAX tree order (dict keys sorted alphabetically)
    const float* ffn_img_fc1_b = F(6);
    const float* ffn_img_fc1_w = F(7);
    const float* ffn_img_fc2_b = F(8);
    const float* ffn_img_fc2_w = F(9);
    const float* ffn_pts_fc1_b = F(10);
    const float* ffn_pts_fc1_w = F(11);
    const float* ffn_pts_fc2_b = F(12);
    const float* ffn_pts_fc2_w = F(13);
    const float* i2p_aw_b  = F(14);
    const float* i2p_aw_w  = F(15);
    const float* i2p_off_b = F(16);
    const float* i2p_off_w = F(17);
    const float* i2p_op_b  = F(18);
    const float* i2p_op_w  = F(19);
    const float* i2p_vp_b  = F(20);
    const float* i2p_vp_w  = F(21);
    const float* ln1_img_b = F(22);
    const float* ln1_img_g = F(23);
    const float* ln1_pts_b = F(24);
    const float* ln1_pts_g = F(25);
    const float* ln2_img_b = F(26);
    const float* ln2_img_g = F(27);
    const float* ln2_pts_b = F(28);
    const float* ln2_pts_g = F(29);
    const float* p2i_aw_b  = F(30);
    const float* p2i_aw_w  = F(31);
    const float* p2i_off_b = F(32);
    const float* p2i_off_w = F(33);
    const float* p2i_op_b  = F(34);
    const float* p2i_op_w  = F(35);
    const float* p2i_vp_b  = F(36);
    const float* p2i_vp_w  = F(37);

    // Workspace layout (floats); buffers reused across the two sequential branches
    float* ws   = (float*)d_ws;
    float* OFFb = ws;                                // NQ*64
    float* AWb  = OFFb + (size_t)NQ * 64;            // NQ*32
    float* Vb   = AWb  + (size_t)NQ * 32;            // NQ*256  (v -> attn -> ffn_out)
    float* Sb   = Vb   + (size_t)NQ * 256;           // NQ*256  (sampled -> x1)
    float* Hb   = Sb   + (size_t)NQ * 256;           // NQ*1024 (ffn hidden)

    const dim3 blk(256);
    const int GM = NQ / 64;               // 625 M-tiles
    const int LN_BLOCKS = NQ / 8;         // wave per row, 8 rows per block

    auto branch = [&](const float* query, const float* value, const float* conf,
                      const float* vp_w, const float* vp_b,
                      const float* off_w, const float* off_b,
                      const float* aw_w, const float* aw_b,
                      const float* op_w, const float* op_b,
                      const float* l1g, const float* l1b,
                      const float* l2g, const float* l2b,
                      const float* fc1w, const float* fc1b,
                      const float* fc2w, const float* fc2b,
                      float* out) {
        // value projection  [NQ,256] = value @ vp_w + vp_b
        gemm_wmma<64, false, false><<<dim3(GM, CC / 64), blk, 0, stream>>>(
            value, vp_w, vp_b, nullptr, Vb, NQ, CC, CC);
        // sampling offsets  [NQ,64]
        gemm_wmma<32, false, false><<<dim3(GM, 64 / 32), blk, 0, stream>>>(
            query, off_w, off_b, nullptr, OFFb, NQ, 64, CC);
        // attention logits  [NQ,32]
        gemm_wmma<32, false, false><<<dim3(GM, 1), blk, 0, stream>>>(
            query, aw_w, aw_b, nullptr, AWb, NQ, 32, CC);
        // confidence-reweighted deformable sampling -> Sb [NQ,256]
        deform_sample<<<dim3(NQ * NHH / 8), blk, 0, stream>>>(Vb, OFFb, AWb, conf, Sb);
        // output projection + identity residual -> Vb
        gemm_wmma<64, false, true><<<dim3(GM, CC / 64), blk, 0, stream>>>(
            Sb, op_w, op_b, query, Vb, NQ, CC, CC);
        // LN1 -> Sb (x1)
        layernorm<<<dim3(LN_BLOCKS), blk, 0, stream>>>(Vb, nullptr, l1g, l1b, Sb, NQ);
        // FFN fc1 + ReLU -> Hb [NQ,1024]
        gemm_wmma<64, true, false><<<dim3(GM, FFND / 64), blk, 0, stream>>>(
            Sb, fc1w, fc1b, nullptr, Hb, NQ, FFND, CC);
        // FFN fc2 -> Vb [NQ,256]
        gemm_wmma<64, false, false><<<dim3(GM, CC / 64), blk, 0, stream>>>(
            Hb, fc2w, fc2b, nullptr, Vb, NQ, CC, FFND);
        // LN2 of (x1 + ffn_out) -> out
        layernorm<<<dim3(LN_BLOCKS), blk, 0, stream>>>(Sb, Vb, l2g, l2b, out, NQ);
    };

    float* out_img = (float*)d_out;
    float* out_pts = out_img + (size_t)NQ * CC;

    // img = deform_attn(i2p, query=img_feat, value=pts_feat, conf=c_pts) -> LN/FFN (img params)
    branch(img_feat, pts_feat, c_pts,
           i2p_vp_w, i2p_vp_b, i2p_off_w, i2p_off_b, i2p_aw_w, i2p_aw_b,
           i2p_op_w, i2p_op_b,
           ln1_img_g, ln1_img_b, ln2_img_g, ln2_img_b,
           ffn_img_fc1_w, ffn_img_fc1_b, ffn_img_fc2_w, ffn_img_fc2_b,
           out_img);

    // pts = deform_attn(p2i, query=pts_feat, value=img_feat, conf=c_img) -> LN/FFN (pts params)
    branch(pts_feat, img_feat, c_img,
           p2i_vp_w, p2i_vp_b, p2i_off_w, p2i_off_b, p2i_aw_w, p2i_aw_b,
           p2i_op_w, p2i_op_b,
           ln1_pts_g, ln1_pts_b, ln2_pts_g, ln2_pts_b,
           ffn_pts_fc1_w, ffn_pts_fc1_b, ffn_pts_fc2_w, ffn_pts_fc2_b,
           out_pts);
}